// Model_41669772706237
// MI455X (gfx1250) — compile-verified
//
#include <hip/hip_runtime.h>
#include <math.h>

typedef __attribute__((ext_vector_type(2))) float v2f;
typedef __attribute__((ext_vector_type(8))) float v8f;

#define KP 516            // padded row length for G (513 data + 3 zeros)
#define MS 1056           // mu row stride (1025 used)

// workspace offsets (in floats)
static constexpr size_t OFF_G   = 0;        // 1025*516 = 528900
static constexpr size_t OFF_POS = 528912;   // 1024
static constexpr size_t OFF_NEG = 529936;   // 1024
static constexpr size_t OFF_WRC = 530960;   // 1024
static constexpr size_t OFF_SC  = 531984;   // 512 scalar slots
static constexpr size_t OFF_MU0 = 532496;   // 64*1056
static constexpr size_t OFF_MU1 = 600080;   // 64*1056
static constexpr size_t OFF_U7  = 667664;   // 64*1024
// sc layout: [0]=scal  [4..5]=c0k  [8..71]=alpha  [72..135]=beta
//            [136..199]=u3cr  [264..327]=rc_vec

__device__ inline v8f wmma4(v2f a, v2f b, v8f c) {
  return __builtin_amdgcn_wmma_f32_16x16x4_f32(false, a, false, b, (short)0, c,
                                               false, false);
}

// ---------------- build normalized G (1025 x 516) ----------------
__global__ __launch_bounds__(256) void build_G(const float* __restrict__ A,
                                               const float* __restrict__ b,
                                               const float* __restrict__ c,
                                               float* __restrict__ G) {
  int r = blockIdx.x;
  int t = threadIdx.x;
  float* grow = G + (size_t)r * KP;
  if (r == 1024) {
    for (int j = t; j < KP; j += 256) grow[j] = (j < 512) ? c[j] : 0.f;
    return;
  }
  const float* arow = A + (size_t)r * 512;
  float ss = 0.f;
  for (int j = t; j < 513; j += 256) {
    float v = (j < 512) ? arow[j] : b[r];
    ss += v * v;
  }
  for (int m = 1; m < 32; m <<= 1) ss += __shfl_xor(ss, m, 32);
  __shared__ float wsum[8];
  __shared__ float inv_s;
  if ((t & 31) == 0) wsum[t >> 5] = ss;
  __syncthreads();
  if (t == 0) {
    float tot = 0.f;
    for (int i = 0; i < 8; ++i) tot += wsum[i];
    float nrm = sqrtf(tot);
    if (nrm < 1e-12f) nrm = 1e-12f;
    inv_s = 1.0f / nrm;
  }
  __syncthreads();
  float inv = inv_s;
  for (int j = t; j < KP; j += 256) {
    float v = (j < 512) ? arow[j] : ((j == 512) ? b[r] : 0.f);
    grow[j] = v * inv;
  }
}

// ------------- wrc[i] = G[i]·G[1024]; also zero pos/neg -------------
__global__ __launch_bounds__(256) void wrc_init(const float* __restrict__ G,
                                                float* __restrict__ wrc,
                                                float* __restrict__ pos,
                                                float* __restrict__ neg) {
  int i = blockIdx.x * 256 + threadIdx.x;
  if (i >= 1024) return;
  pos[i] = 0.f;
  neg[i] = 0.f;
  const float* gi = G + (size_t)i * KP;
  const float* gc = G + (size_t)1024 * KP;
  float d = 0.f;
  for (int k = 0; k < 513; ++k) d += gi[k] * gc[k];
  wrc[i] = d;
}

// --- pairwise: W tile = G_i G_jᵀ via WMMA; reduce relu± into pos/neg ---
__global__ __launch_bounds__(256) void w_pair(const float* __restrict__ G,
                                              float* __restrict__ pos,
                                              float* __restrict__ neg) {
  int bi = blockIdx.x, bj = blockIdx.y;
  int w = threadIdx.x >> 5;       // wave 0..7
  int lane = threadIdx.x & 31;
  int ti = w >> 1;                // 0..3
  int tjb = (w & 1) * 2;          // 0 or 2
  int i0 = bi * 64 + ti * 16;
  int j0 = bj * 64 + tjb * 16;
  int m16 = lane & 15;
  int half = lane >> 4;
  int kh = half * 2;
  const float* arow = G + (size_t)(i0 + m16) * KP + kh;
  const float* brow0 = G + (size_t)(j0 + m16) * KP + kh;
  const float* brow1 = brow0 + (size_t)16 * KP;
  v8f c0 = {};
  v8f c1 = {};
  for (int k0 = 0; k0 < KP; k0 += 4) {
    v2f a = *(const v2f*)(arow + k0);
    v2f b0 = *(const v2f*)(brow0 + k0);
    v2f b1 = *(const v2f*)(brow1 + k0);
    c0 = wmma4(a, b0, c0);
    c1 = wmma4(a, b1, c1);
  }
  int gj = j0 + m16;  // this lane's column (tile 0); tile 1 = gj+16
#pragma unroll
  for (int r = 0; r < 8; ++r) {
    int gi = i0 + r + 8 * half;
    float v = c0[r];
    if (gi == gj) v = 0.f;  // zero diagonal of W
    float p = fmaxf(v, 0.f), n = fminf(v, 0.f);
    for (int m = 1; m < 16; m <<= 1) {
      p += __shfl_xor(p, m, 32);
      n += __shfl_xor(n, m, 32);
    }
    if (m16 == 0) {
      atomicAdd(&pos[gi], p);
      atomicAdd(&neg[gi], n);
    }
    v = c1[r];
    if (gi == gj + 16) v = 0.f;
    p = fmaxf(v, 0.f);
    n = fminf(v, 0.f);
    for (int m = 1; m < 16; m <<= 1) {
      p += __shfl_xor(p, m, 32);
      n += __shfl_xor(n, m, 32);
    }
    if (m16 == 0) {
      atomicAdd(&pos[gi], p);
      atomicAdd(&neg[gi], n);
    }
  }
}

// --- iteration-invariant small terms: alpha, beta, u3cr, scal ---
__global__ void precompute_k(const float* __restrict__ wrc,
                             const float* __restrict__ t3rr,
                             const float* __restrict__ t3cr,
                             const float* __restrict__ t4rr,
                             const float* __restrict__ t4rc,
                             const float* __restrict__ t4cr,
                             float* __restrict__ sc) {
  __shared__ float t4v[64], rp[64], rn[64];
  __shared__ float posW, negW;
  int q = threadIdx.x;
  t4v[q] = t4rr[q];
  float p = 0.f, n = 0.f;
  for (int j = q; j < 1024; j += 64) {
    float w = wrc[j];
    p += fmaxf(w, 0.f);
    n += fminf(w, 0.f);
  }
  rp[q] = p;
  rn[q] = n;
  __syncthreads();
  if (q == 0) {
    float tp = 0.f, tn = 0.f;
    for (int i = 0; i < 64; ++i) { tp += rp[i]; tn += rn[i]; }
    posW = tp;
    negW = tn;
    float s = 0.f;
    for (int i = 0; i < 64; ++i) {
      float t = t4rc[i];
      if (t > 0.f) s += t * t * tp;
      else if (t < 0.f) s += t * t * tn;
    }
    sc[0] = s;  // scal
  }
  __syncthreads();
  float a = 0.f, bb = 0.f;
  for (int i = 0; i < 64; ++i) {
    float t = t4v[i];
    float w = t3rr[q * 64 + i];
    if (t > 0.f) a += w * t;
    else if (t < 0.f) bb += w * t;
  }
  sc[8 + q] = a;
  sc[72 + q] = bb;
  float u3 = 0.f;
  for (int i = 0; i < 64; ++i) {
    float t = t4cr[i];
    float rc = (t > 0.f) ? t * posW : ((t < 0.f) ? t * negW : 0.f);
    u3 += t3cr[q * 64 + i] * rc;
  }
  sc[136 + q] = u3;
}

__global__ __launch_bounds__(256) void zero_mu(float* __restrict__ mu) {
  int i = blockIdx.x * 256 + threadIdx.x;
  if (i < 64 * MS) mu[i] = 0.f;
}

// --- per-iteration scalar terms + column m of new mu ---
__global__ void iter_small(const float* __restrict__ src, float* __restrict__ dst,
                           float* __restrict__ sc, const float* __restrict__ t2rc,
                           const float* __restrict__ t2cr,
                           const float* __restrict__ t0,
                           const float* __restrict__ t1,
                           const float* __restrict__ nf) {
  __shared__ float colsum[64], mucol[64];
  int q = threadIdx.x;
  float s = 0.f;
  const float* row = src + (size_t)q * MS;
  for (int j = 0; j < 1024; ++j) s += row[j];
  colsum[q] = s;
  mucol[q] = row[1024];
  __syncthreads();
  const float Cm = 1.0f / 1024.0f;
  float rc = 0.f, u2c = 0.f;
  for (int p = 0; p < 64; ++p) {
    rc += t2rc[q * 64 + p] * mucol[p];
    u2c += t2cr[q * 64 + p] * (Cm * colsum[p]);
  }
  sc[264 + q] = rc;  // rc_vec broadcast term for columns < m
  float val = t0[q] + t1[q] * nf[1024] + u2c + sc[136 + q];
  dst[(size_t)q * MS + 1024] = fmaxf(val, 0.f);
}

// --- main recurrence: dst[:, :1024] = relu(t2rr@src + rank2 + bcast) ---
__global__ __launch_bounds__(256) void iter_gemm(const float* __restrict__ t2rr,
                                                 const float* __restrict__ src,
                                                 float* __restrict__ dst,
                                                 const float* __restrict__ t0,
                                                 const float* __restrict__ t1,
                                                 const float* __restrict__ nf,
                                                 const float* __restrict__ pos,
                                                 const float* __restrict__ neg,
                                                 const float* __restrict__ sc) {
  int w = blockIdx.x * 8 + (threadIdx.x >> 5);
  int lane = threadIdx.x & 31;
  int qt = w & 3, it = w >> 2;
  int q0 = qt * 16, i0 = it * 16;
  int m16 = lane & 15, half = lane >> 4, kh = half * 2;
  const float* arow = t2rr + (q0 + m16) * 64 + kh;
  v8f c = {};
  for (int k0 = 0; k0 < 64; k0 += 4) {
    v2f a = *(const v2f*)(arow + k0);
    v2f b;
    b.x = src[(size_t)(k0 + kh) * MS + i0 + m16];
    b.y = src[(size_t)(k0 + kh + 1) * MS + i0 + m16];
    c = wmma4(a, b, c);
  }
  float scal = sc[0];
  int gi = i0 + m16;
  float nfv = nf[gi], ps = pos[gi], ns = neg[gi];
#pragma unroll
  for (int r = 0; r < 8; ++r) {
    int gq = q0 + r + 8 * half;
    float val = c[r] + t0[gq] + t1[gq] * nfv + sc[8 + gq] * ps +
                sc[72 + gq] * ns + scal + sc[264 + gq];
    dst[(size_t)gq * MS + gi] = fmaxf(val, 0.f);
  }
}

// --- final small terms: u6, class constants c0k ---
__global__ void final_small(const float* __restrict__ mu,
                            const float* __restrict__ t6r,
                            const float* __restrict__ t6c,
                            const float* __restrict__ t8,
                            float* __restrict__ sc) {
  __shared__ float colsum[64], mucol[64], u6r[64];
  int q = threadIdx.x;
  float s = 0.f;
  const float* row = mu + (size_t)q * MS;
  for (int j = 0; j < 1024; ++j) s += row[j];
  colsum[q] = s;
  mucol[q] = row[1024];
  __syncthreads();
  const float Cm = 1.0f / 1024.0f;
  float u6 = 0.f;
  for (int p = 0; p < 64; ++p)
    u6 += t6r[q * 64 + p] * (Cm * colsum[p]) + t6c[q * 64 + p] * mucol[p];
  u6r[q] = fmaxf(u6, 0.f);
  __syncthreads();
  if (q < 2) {
    float cv = 0.f;
    for (int r = 0; r < 64; ++r) cv += u6r[r] * t8[q * 128 + r];
    sc[4 + q] = cv;
  }
}

// --- relu(u7) = relu(t7 @ mu[:, in_loss]) via WMMA ---
__global__ __launch_bounds__(256) void u7_gemm(const float* __restrict__ t7,
                                               const float* __restrict__ mu,
                                               const int* __restrict__ in_loss,
                                               float* __restrict__ ru7) {
  int w = blockIdx.x * 8 + (threadIdx.x >> 5);
  int lane = threadIdx.x & 31;
  int qt = w & 3, it = w >> 2;
  int q0 = qt * 16, i0 = it * 16;
  int m16 = lane & 15, half = lane >> 4, kh = half * 2;
  int col = in_loss[i0 + m16];  // gathered mu column for this lane
  const float* arow = t7 + (q0 + m16) * 64 + kh;
  v8f c = {};
  for (int k0 = 0; k0 < 64; k0 += 4) {
    v2f a = *(const v2f*)(arow + k0);
    v2f b;
    b.x = mu[(size_t)(k0 + kh) * MS + col];
    b.y = mu[(size_t)(k0 + kh + 1) * MS + col];
    c = wmma4(a, b, c);
  }
  int gi = i0 + m16;
#pragma unroll
  for (int r = 0; r < 8; ++r) {
    int gq = q0 + r + 8 * half;
    ru7[(size_t)gq * 1024 + gi] = fmaxf(c[r], 0.f);
  }
}

// --- scores + 2-class log_softmax ---
__global__ __launch_bounds__(256) void score_k(const float* __restrict__ ru7,
                                               const float* __restrict__ t8,
                                               const float* __restrict__ sc,
                                               float* __restrict__ out) {
  int l = blockIdx.x * 256 + threadIdx.x;
  if (l >= 1024) return;
  float s0 = sc[4], s1 = sc[5];
  for (int q = 0; q < 64; ++q) {
    float f = ru7[(size_t)q * 1024 + l];
    s0 += f * t8[64 + q];
    s1 += f * t8[128 + 64 + q];
  }
  float mx = fmaxf(s0, s1);
  float lse = mx + logf(expf(s0 - mx) + expf(s1 - mx));
  out[2 * l + 0] = s0 - lse;
  out[2 * l + 1] = s1 - lse;
}

extern "C" void kernel_launch(void* const* d_in, const int* in_sizes, int n_in,
                              void* d_out, int out_size, void* d_ws,
                              size_t ws_size, hipStream_t stream) {
  const float* A = (const float*)d_in[0];
  const float* b = (const float*)d_in[1];
  const float* c = (const float*)d_in[2];
  const float* nf = (const float*)d_in[3];
  const int* in_loss = (const int*)d_in[4];
  const float* t0 = (const float*)d_in[5];
  const float* t1 = (const float*)d_in[6];
  const float* t2rr = (const float*)d_in[7];
  const float* t2rc = (const float*)d_in[8];
  const float* t2cr = (const float*)d_in[9];
  const float* t3rr = (const float*)d_in[10];
  const float* t3cr = (const float*)d_in[11];
  const float* t4rr = (const float*)d_in[12];
  const float* t4rc = (const float*)d_in[13];
  const float* t4cr = (const float*)d_in[14];
  const float* t6r = (const float*)d_in[15];
  const float* t6c = (const float*)d_in[16];
  const float* t7 = (const float*)d_in[17];
  const float* t8 = (const float*)d_in[18];

  float* ws = (float*)d_ws;
  float* G = ws + OFF_G;
  float* pos = ws + OFF_POS;
  float* neg = ws + OFF_NEG;
  float* wrc = ws + OFF_WRC;
  float* sc = ws + OFF_SC;
  float* mu0 = ws + OFF_MU0;
  float* mu1 = ws + OFF_MU1;
  float* ru7 = ws + OFF_U7;

  build_G<<<1025, 256, 0, stream>>>(A, b, c, G);
  wrc_init<<<4, 256, 0, stream>>>(G, wrc, pos, neg);
  w_pair<<<dim3(16, 16), 256, 0, stream>>>(G, pos, neg);
  precompute_k<<<1, 64, 0, stream>>>(wrc, t3rr, t3cr, t4rr, t4rc, t4cr, sc);
  zero_mu<<<(64 * MS + 255) / 256, 256, 0, stream>>>(mu0);

  float* src = mu0;
  float* dst = mu1;
  for (int t = 0; t < 3; ++t) {
    iter_small<<<1, 64, 0, stream>>>(src, dst, sc, t2rc, t2cr, t0, t1, nf);
    iter_gemm<<<32, 256, 0, stream>>>(t2rr, src, dst, t0, t1, nf, pos, neg, sc);
    float* tmp = src; src = dst; dst = tmp;
  }
  // final mu is in `src` (mu1 after 3 swaps)
  final_small<<<1, 64, 0, stream>>>(src, t6r, t6c, t8, sc);
  u7_gemm<<<32, 256, 0, stream>>>(t7, src, in_loss, ru7);
  score_k<<<4, 256, 0, stream>>>(ru7, t8, sc, (float*)d_out);
}